// e2edro_nn_64974265254414
// MI455X (gfx1250) — compile-verified
//
#include <hip/hip_runtime.h>
#include <hip/hip_bf16.h>

#define N_Xd   30
#define N_Yd   50
#define N_OBS  256
#define ITERS  400
#define LRATE  0.05f
#define KPAD   52          // N_Y padded to multiple of 4 for WMMA K-steps
#define SPB    16          // scenarios per block

typedef float v2f __attribute__((ext_vector_type(2)));
typedef float v8f __attribute__((ext_vector_type(8)));

__device__ __forceinline__ v8f wmma_f32(v2f a, v2f b, v8f c) {
  // D = A(16x4,f32) * B(4x16,f32) + C(16x16,f32), wave32
  return __builtin_amdgcn_wmma_f32_16x16x4_f32(false, a, false, b,
                                               (short)0, c, false, false);
}

__global__ __launch_bounds__(256)
void e2edro_kernel(const float* __restrict__ X, const float* __restrict__ Y,
                   const float* __restrict__ W, const float* __restrict__ b,
                   const float* __restrict__ deltap, const float* __restrict__ gammap,
                   float* __restrict__ out)
{
  __shared__ float s_ep[N_OBS][KPAD];   // ep = Y - Yhat, K-padded with zeros
  __shared__ float s_S [N_OBS][SPB];    // S = ep @ Z
  __shared__ float s_Q [N_OBS][SPB];    // Q = coef * 2*(S - c)
  __shared__ float s_Z [KPAD][SPB];     // current z per scenario (column-major)
  __shared__ float s_yh[SPB][KPAD];     // Y_hat rows for this block's scenarios
  __shared__ float s_GZ[64][SPB];       // GZ = ep^T @ Q ; reused as V
  __shared__ float s_U [64][SPB];       // descending-sorted V (scatter by rank)
  __shared__ float s_red[16][SPB];      // gc reduction partials
  __shared__ float s_c  [SPB];
  __shared__ float s_tau[SPB];

  const int tid  = threadIdx.x;
  const int wg   = blockIdx.x;
  const int wave = tid >> 5;
  const int lane = tid & 31;
  const int half = lane >> 4;
  const int l16  = lane & 15;

  const float delta = deltap[0];
  const float gamma = gammap[0];
  const float aa    = fminf(delta * 0.5f, 255.0f / 256.0f);

  // ---------------- init: Y_hat (full), ep, own-yhat, Z, c ----------------
  {
    const int i = tid;                       // one observation row per thread
    float xr[N_Xd];
    #pragma unroll
    for (int k = 0; k < N_Xd; ++k) xr[k] = X[i * N_Xd + k];
    for (int j = 0; j < N_Yd; ++j) {
      float acc = b[j];
      #pragma unroll
      for (int k = 0; k < N_Xd; ++k) acc += xr[k] * W[j * N_Xd + k];
      s_ep[i][j] = Y[i * N_Yd + j] - acc;
      if ((i >> 4) == wg) {
        s_yh[i & 15][j] = acc;
        out[N_OBS * N_Yd + i * N_Yd + j] = acc;     // Y_hat output (owned rows)
      }
    }
    s_ep[i][50] = 0.f; s_ep[i][51] = 0.f;
    for (int idx = tid; idx < KPAD * SPB; idx += 256) {
      const int j = idx >> 4, n = idx & 15;
      s_Z[j][n] = (j < N_Yd) ? (1.0f / (float)N_Yd) : 0.f;
    }
    if (tid < SPB) s_c[tid] = 0.f;
  }
  __syncthreads();

  for (int it = 0; it < ITERS; ++it) {
    // ---- S = ep @ Z : 16 M-tiles x 1 N-tile, K=52 (13 wmma k-steps) ----
    #pragma unroll
    for (int t = 0; t < 2; ++t) {
      const int m0 = (wave + 8 * t) << 4;
      v8f acc = {};
      #pragma unroll
      for (int k0 = 0; k0 < KPAD; k0 += 4) {
        const int ka = k0 + 2 * half;
        v2f av, bv;
        av.x = s_ep[m0 + l16][ka];
        av.y = s_ep[m0 + l16][ka + 1];
        bv.x = s_Z[ka][l16];
        bv.y = s_Z[ka + 1][l16];
        acc = wmma_f32(av, bv, acc);
      }
      #pragma unroll
      for (int v = 0; v < 8; ++v)
        s_S[m0 + v + 8 * half][l16] = acc[v];
    }
    __syncthreads();

    // ---- count-rank on |S - c| -> worst_case VJP coef -> Q ----
    {
      const int scen = tid & 15;
      const int m    = tid >> 4;
      const float cs = s_c[scen];
      float dv[16], xa[16]; int cnt[16];
      #pragma unroll
      for (int r = 0; r < 16; ++r) {
        const int i = m + (r << 4);
        dv[r] = s_S[i][scen] - cs;
        xa[r] = fabsf(dv[r]);
        cnt[r] = 0;
      }
      for (int j = 0; j < N_OBS; ++j) {
        const float xj = fabsf(s_S[j][scen] - cs);
        #pragma unroll
        for (int r = 0; r < 16; ++r) {
          const int i = m + (r << 4);
          cnt[r] += (int)((xj < xa[r]) || (xj == xa[r] && j < i));
        }
      }
      #pragma unroll
      for (int r = 0; r < 16; ++r) {
        const int i = m + (r << 4);
        const float w = fminf(fmaxf(aa - (float)cnt[r] * (1.0f / 256.0f), 0.0f),
                              1.0f / 256.0f);
        float coef = (1.0f / 256.0f) - w;
        if (cnt[r] == N_OBS - 1) coef += aa;     // d(a*max(r))/dr at argmax
        s_Q[i][scen] = coef * 2.0f * dv[r];
      }
    }
    __syncthreads();

    // ---- gc partials (all threads) ; GZ = ep^T @ Q (waves 0..3) ----
    {
      const int scen = tid & 15;
      const int m    = tid >> 4;
      float p = 0.f;
      #pragma unroll
      for (int r = 0; r < 16; ++r) p += s_Q[(m << 4) + r][scen];
      s_red[m][scen] = p;
    }
    if (wave < 4) {
      const int m0 = wave << 4;
      const int m  = m0 + l16;
      v8f acc = {};
      #pragma unroll 16
      for (int k0 = 0; k0 < N_OBS; k0 += 4) {
        const int ka = k0 + 2 * half;
        v2f av, bv;
        av.x = (m < KPAD) ? s_ep[ka][m]     : 0.f;   // A = ep^T (transposed read)
        av.y = (m < KPAD) ? s_ep[ka + 1][m] : 0.f;
        bv.x = s_Q[ka][l16];
        bv.y = s_Q[ka + 1][l16];
        acc = wmma_f32(av, bv, acc);
      }
      #pragma unroll
      for (int v = 0; v < 8; ++v)
        s_GZ[m0 + v + 8 * half][l16] = acc[v];
    }
    __syncthreads();

    // ---- c update ; V = Z - LR*(GZ - gamma*yhat) (in place over GZ) ----
    if (tid < SPB) {
      float g = 0.f;
      #pragma unroll
      for (int m = 0; m < 16; ++m) g += s_red[m][tid];
      s_c[tid] += LRATE * g;                 // c -= LR*gc, gc = -sum(Q col)
    }
    {
      const int scen = tid & 15;
      const int m    = tid >> 4;
      for (int j = m; j < N_Yd; j += 16) {
        const float gz = s_GZ[j][scen] - gamma * s_yh[scen][j];
        s_GZ[j][scen] = s_Z[j][scen] - LRATE * gz;
      }
    }
    __syncthreads();

    // ---- descending count-rank of V (len 50), scatter sorted into U ----
    {
      const int scen = tid & 15;
      const int m    = tid >> 4;
      float vv[4]; int jj[4]; int cnt[4];
      #pragma unroll
      for (int q = 0; q < 4; ++q) {
        const int j = m + (q << 4);
        jj[q]  = j;
        vv[q]  = (j < N_Yd) ? s_GZ[j][scen] : -1e30f;
        cnt[q] = 0;
      }
      for (int k = 0; k < N_Yd; ++k) {
        const float vk = s_GZ[k][scen];
        #pragma unroll
        for (int q = 0; q < 4; ++q)
          cnt[q] += (int)((vk > vv[q]) || (vk == vv[q] && k < jj[q]));
      }
      #pragma unroll
      for (int q = 0; q < 4; ++q)
        if (jj[q] < N_Yd) s_U[cnt[q]][scen] = vv[q];
    }
    __syncthreads();

    // ---- simplex threshold tau per scenario ----
    if (tid < SPB) {
      float css = -1.0f, best = 0.f; int rho = 1;
      for (int k = 0; k < N_Yd; ++k) {
        const float u = s_U[k][tid];
        css += u;
        if (u - css / (float)(k + 1) > 0.f) { rho = k + 1; best = css; }
      }
      s_tau[tid] = best / (float)rho;
    }
    __syncthreads();

    // ---- z = max(V - tau, 0) ----
    {
      const int scen = tid & 15;
      const int m    = tid >> 4;
      const float tv = s_tau[scen];
      for (int j = m; j < N_Yd; j += 16)
        s_Z[j][scen] = fmaxf(s_GZ[j][scen] - tv, 0.f);
    }
    __syncthreads();
  }

  // ---------------- write Z_star ----------------
  for (int idx = tid; idx < SPB * N_Yd; idx += 256) {
    const int tl = idx / N_Yd, j = idx % N_Yd;
    out[(wg * SPB + tl) * N_Yd + j] = s_Z[j][tl];
  }
}

extern "C" void kernel_launch(void* const* d_in, const int* in_sizes, int n_in,
                              void* d_out, int out_size, void* d_ws, size_t ws_size,
                              hipStream_t stream) {
  const float* X  = (const float*)d_in[0];
  const float* Y  = (const float*)d_in[1];
  const float* W  = (const float*)d_in[2];
  const float* b  = (const float*)d_in[3];
  const float* dl = (const float*)d_in[4];
  const float* gm = (const float*)d_in[5];
  float* out = (float*)d_out;
  e2edro_kernel<<<dim3(N_OBS / SPB), dim3(256), 0, stream>>>(X, Y, W, b, dl, gm, out);
}